// LossFunction_18090402251320
// MI455X (gfx1250) — compile-verified
//
#include <hip/hip_runtime.h>
#include <hip/hip_bf16.h>

typedef __attribute__((ext_vector_type(2))) float v2f;
typedef __attribute__((ext_vector_type(8))) float v8f;

#define NB      8192      // batch rows
#define DIM     512       // feature dim
#define EPSN    1e-12f    // norm epsilon
#define EPSPD   1e-6f     // pairwise-distance epsilon
#define MARG    0.3f

// ---------------------------------------------------------------------------
// Kernel 1: row-normalize a (sel=0) and p (sel=1); emit per-column argmax
// constant c[j] = -sp2[j] + 2*eps*sp[j] for the fused d2 score.
// ---------------------------------------------------------------------------
__global__ __launch_bounds__(256) void normalize_kernel(const float* __restrict__ x,
                                                        float* __restrict__ an,
                                                        float* __restrict__ pn,
                                                        float* __restrict__ cvec)
{
    __shared__ float rss[256];
    __shared__ float rs[256];
    const int tid = threadIdx.x;
    const int row = blockIdx.x >> 1;
    const int sel = blockIdx.x & 1;

    const float2 v = ((const float2*)(x + (size_t)row * (2 * DIM) + sel * DIM))[tid];
    rss[tid] = v.x * v.x + v.y * v.y;
    rs[tid]  = v.x + v.y;
    __syncthreads();
    for (int off = 128; off > 0; off >>= 1) {
        if (tid < off) { rss[tid] += rss[tid + off]; rs[tid] += rs[tid + off]; }
        __syncthreads();
    }
    const float tot_ss = rss[0];
    const float tot_s  = rs[0];
    const float inv = 1.0f / fmaxf(sqrtf(tot_ss), EPSN);

    float* dst = (sel ? pn : an) + (size_t)row * DIM;
    float2 o; o.x = v.x * inv; o.y = v.y * inv;
    ((float2*)dst)[tid] = o;

    if (sel == 1 && tid == 0) {
        const float sp2 = tot_ss * inv * inv;   // sum(p_norm^2)
        const float sp  = tot_s * inv;          // sum(p_norm)
        cvec[row] = -sp2 + 2.0f * EPSPD * sp;
    }
}

// ---------------------------------------------------------------------------
// Kernel 2: fused  dot = a @ p.T  +  row-argmax of sim (hardest negative).
// One block = 16 output rows; A-tile (16x512 f32, 32 KB) staged in LDS and
// shared by 8 wave32s; each wave sweeps 16-column tiles with
// V_WMMA_F32_16X16X4_F32 (K=4, 128 WMMAs per tile).
// ---------------------------------------------------------------------------
__global__ __launch_bounds__(256) void neg_argmax_kernel(const float* __restrict__ an,
                                                         const float* __restrict__ pn,
                                                         const float* __restrict__ cvec,
                                                         int* __restrict__ negidx)
{
    __shared__ float Als[16 * DIM];        // 32 KB A tile
    __shared__ float redv[8 * 32 * 8];     // per-(wave,lane,slot) best value
    __shared__ int   redi[8 * 32 * 8];     // per-(wave,lane,slot) best index

    const int tid = threadIdx.x;
    const int rowbase = blockIdx.x * 16;

    // Stage A tile: 16*512 floats = 2048 float4, 8 per thread.
    {
        const float4* src = (const float4*)(an + (size_t)rowbase * DIM);
        float4* dst = (float4*)Als;
        #pragma unroll
        for (int i = 0; i < 8; ++i)
            dst[tid + i * 256] = src[tid + i * 256];
    }
    __syncthreads();

    const int lane = tid & 31;
    const int wave = tid >> 5;
    const int half = lane >> 4;   // K-half selector for A/B fragments
    const int l16  = lane & 15;   // M (for A) / N (for B) within tile
    const float* aptr = Als + l16 * DIM + half * 2;

    float best[8];
    int   bidx[8];
    #pragma unroll
    for (int r = 0; r < 8; ++r) { best[r] = -__builtin_inff(); bidx[r] = 0x7fffffff; }

    for (int jt = wave; jt < (NB / 16); jt += 8) {
        const int colbase = jt * 16;
        const float* bptr = pn + (size_t)(colbase + l16) * DIM + half * 2;

        v8f acc = {0.f, 0.f, 0.f, 0.f, 0.f, 0.f, 0.f, 0.f};
        #pragma unroll 8
        for (int k = 0; k < DIM; k += 4) {
            const v2f af = *(const v2f*)(aptr + k);   // ds_load_b64
            const v2f bf = *(const v2f*)(bptr + k);   // global_load_b64
            acc = __builtin_amdgcn_wmma_f32_16x16x4_f32(
                false, af, false, bf, (short)0, acc, false, false);
        }

        const int   col = colbase + l16;
        const float c   = cvec[col];
        #pragma unroll
        for (int r = 0; r < 8; ++r) {
            const int row = rowbase + half * 8 + r;   // C-layout: VGPR r, lane half
            float sv = 2.0f * acc[r] + c;             // monotone in sim = -d2
            if (row == col) sv = -__builtin_inff();   // mask diagonal
            if (sv > best[r]) { best[r] = sv; bidx[r] = col; }
        }
    }

    const int base = tid * 8;
    #pragma unroll
    for (int r = 0; r < 8; ++r) { redv[base + r] = best[r]; redi[base + r] = bidx[r]; }
    __syncthreads();

    // Cross-wave/lane reduction: row (h*8 + r) lives in slot r of lanes h*16..h*16+15.
    if (tid < 16) {
        const int r = tid & 7;
        const int h = tid >> 3;
        float bv = -__builtin_inff();
        int   bi = 0x7fffffff;
        for (int w = 0; w < 8; ++w) {
            for (int L = 0; L < 16; ++L) {
                const int idx = ((w * 32) + (h * 16) + L) * 8 + r;
                const float v = redv[idx];
                const int   ii = redi[idx];
                if (v > bv || (v == bv && ii < bi)) { bv = v; bi = ii; }  // first-max tie-break
            }
        }
        negidx[rowbase + tid] = bi;
    }
}

// ---------------------------------------------------------------------------
// Kernel 3: per-row hinge loss  relu(||a-p+eps||^2 - ||a-n+eps||^2 + margin)
// ---------------------------------------------------------------------------
__global__ __launch_bounds__(256) void rowloss_kernel(const float* __restrict__ an,
                                                      const float* __restrict__ pn,
                                                      const int* __restrict__ negidx,
                                                      float* __restrict__ rowloss)
{
    __shared__ float rp[256];
    __shared__ float rn[256];
    const int tid = threadIdx.x;
    const int row = blockIdx.x;

    const float2 av = ((const float2*)(an + (size_t)row * DIM))[tid];
    const float2 pv = ((const float2*)(pn + (size_t)row * DIM))[tid];
    const float2 nv = ((const float2*)(pn + (size_t)negidx[row] * DIM))[tid];

    const float dpx = av.x - pv.x + EPSPD, dpy = av.y - pv.y + EPSPD;
    const float dnx = av.x - nv.x + EPSPD, dny = av.y - nv.y + EPSPD;
    rp[tid] = dpx * dpx + dpy * dpy;
    rn[tid] = dnx * dnx + dny * dny;
    __syncthreads();
    for (int off = 128; off > 0; off >>= 1) {
        if (tid < off) { rp[tid] += rp[tid + off]; rn[tid] += rn[tid + off]; }
        __syncthreads();
    }
    if (tid == 0) rowloss[row] = fmaxf(rp[0] - rn[0] + MARG, 0.0f);
}

// ---------------------------------------------------------------------------
// Kernel 4: deterministic fixed-order mean (no float atomics).
// ---------------------------------------------------------------------------
__global__ __launch_bounds__(256) void mean_kernel(const float* __restrict__ rowloss,
                                                   float* __restrict__ out)
{
    __shared__ float red[256];
    const int tid = threadIdx.x;
    float s = 0.0f;
    for (int i = tid; i < NB; i += 256) s += rowloss[i];
    red[tid] = s;
    __syncthreads();
    for (int off = 128; off > 0; off >>= 1) {
        if (tid < off) red[tid] += red[tid + off];
        __syncthreads();
    }
    if (tid == 0) out[0] = red[0] * (1.0f / (float)NB);
}

extern "C" void kernel_launch(void* const* d_in, const int* in_sizes, int n_in,
                              void* d_out, int out_size, void* d_ws, size_t ws_size,
                              hipStream_t stream)
{
    (void)in_sizes; (void)n_in; (void)out_size; (void)ws_size;
    const float* x = (const float*)d_in[0];

    float* an      = (float*)d_ws;                         // NB*DIM
    float* pn      = an + (size_t)NB * DIM;                // NB*DIM
    float* cvec    = pn + (size_t)NB * DIM;                // NB
    int*   negidx  = (int*)(cvec + NB);                    // NB
    float* rowloss = (float*)(negidx + NB);                // NB
    float* out     = (float*)d_out;

    normalize_kernel<<<2 * NB, 256, 0, stream>>>(x, an, pn, cvec);
    neg_argmax_kernel<<<NB / 16, 256, 0, stream>>>(an, pn, cvec, negidx);
    rowloss_kernel<<<NB, 256, 0, stream>>>(an, pn, negidx, rowloss);
    mean_kernel<<<1, 256, 0, stream>>>(rowloss, out);
}